// MemoryEfficientSVDPlaneProjection_2216203125202
// MI455X (gfx1250) — compile-verified
//
#include <hip/hip_runtime.h>
#include <math.h>

#define NPTS   262144
#define NPL    64
#define NBATCH 8
#define THRESH 0.01f

typedef __attribute__((ext_vector_type(2))) float v2f;
typedef __attribute__((ext_vector_type(8))) float v8f;

static __device__ __forceinline__ v8f wmma_dist(v2f a, v2f b) {
    v8f zero = {};
    return __builtin_amdgcn_wmma_f32_16x16x4_f32(
        false, a, false, b, (short)0, zero, false, false);
}

// ---------------------------------------------------------------------------
// Kernel 1: per-plane masked moment reduction, WMMA distance engine.
// Block = 256 threads (8 waves). Wave -> (plane group g = w&3, point chunk
// s = w>>2). Per 32-point step: 2x v_wmma_f32_16x16x4_f32 produce all
// 32 pts x 16 plane distances; per-row ballots gate the rare (~16%) masked
// 10-FMA accumulation. Lane accumulates plane g*16+(lane&15); lane halves
// keep independent partials (slot = blk*4 + s*2 + half), reduced in kernel 2.
// ---------------------------------------------------------------------------
__global__ void __launch_bounds__(256)
k_plane_stats(const float* __restrict__ pts, const float* __restrict__ planes,
              float* __restrict__ partial, int bp, int iters)
{
    const int b    = blockIdx.x / bp;
    const int blk  = blockIdx.x % bp;
    const int lane = threadIdx.x & 31;
    const int wave = threadIdx.x >> 5;
    const int g    = wave & 3;             // plane group (16 planes)
    const int s    = wave >> 2;            // point sub-chunk (0/1)
    const int m    = lane & 15;
    const int h    = lane >> 4;
    const bool lo  = lane < 16;

    // B fragment for this wave's plane group (original unit normal + raw d)
    const float* pp = planes + ((size_t)b * NPL + g * 16 + m) * 4;
    float nx = pp[0], ny = pp[1], nz = pp[2], d = pp[3];
    float nn  = sqrtf(nx * nx + ny * ny + nz * nz);
    float inv = 1.0f / fmaxf(nn, 1e-12f);
    float ux = nx * inv, uy = ny * inv, uz = nz * inv;
    v2f Bg;
    Bg.x = lo ? ux : uz;
    Bg.y = lo ? uy : d;

    float cnt = 0.f, sx = 0.f, sy = 0.f, sz = 0.f;
    float xx = 0.f, xy = 0.f, xz = 0.f, yy = 0.f, yz = 0.f, zz = 0.f;

    const int base0 = blk * (iters * 64) + s * (iters * 32);
    const float* P = pts + ((size_t)b * NPTS + base0) * 3;

    for (int it = 0; it < iters; ++it) {
        const float* Pi = P + (size_t)it * 96;          // 32 points * 3
        float px = Pi[3 * lane], py = Pi[3 * lane + 1], pz = Pi[3 * lane + 2];

        // A fragments: 16x4 fp32 (lanes 0-15: K=0,1 ; lanes 16-31: K=2,3)
        float x0 = __shfl(px, m, 32),      y0 = __shfl(py, m, 32),      z0 = __shfl(pz, m, 32);
        float x1 = __shfl(px, m + 16, 32), y1 = __shfl(py, m + 16, 32), z1 = __shfl(pz, m + 16, 32);
        v2f A0, A1;
        A0.x = lo ? x0 : z0;  A0.y = lo ? y0 : 1.0f;
        A1.x = lo ? x1 : z1;  A1.y = lo ? y1 : 1.0f;

        v8f D0 = wmma_dist(A0, Bg);
        v8f D1 = wmma_dist(A1, Bg);

#pragma unroll
        for (int t = 0; t < 2; ++t) {
            v8f D = t ? D1 : D0;
#pragma unroll
            for (int j = 0; j < 8; ++j) {
                bool hit = fabsf(D[j]) < THRESH;        // lane: plane g*16+m,
                if (__any(hit)) {                       //       point t*16+j+8*h
                    int src = t * 16 + j + (h << 3);
                    float cxp = __shfl(px, src, 32);
                    float cyp = __shfl(py, src, 32);
                    float czp = __shfl(pz, src, 32);
                    float w  = hit ? 1.0f : 0.0f;
                    float wx = w * cxp, wy = w * cyp, wz = w * czp;
                    cnt += w;
                    sx += wx; sy += wy; sz += wz;
                    xx = fmaf(wx, cxp, xx); xy = fmaf(wx, cyp, xy); xz = fmaf(wx, czp, xz);
                    yy = fmaf(wy, cyp, yy); yz = fmaf(wy, czp, yz); zz = fmaf(wz, czp, zz);
                }
            }
        }
    }

    const int plane = g * 16 + m;
    const int slot  = blk * 4 + s * 2 + h;
    float* o = partial + (((size_t)b * (bp * 4) + slot) * NPL + plane) * 10;
    o[0] = cnt; o[1] = sx; o[2] = sy; o[3] = sz; o[4] = xx;
    o[5] = xy;  o[6] = xz; o[7] = yy; o[8] = yz; o[9] = zz;
}

// ---------------------------------------------------------------------------
// Kernel 2: reduce partials, covariance, smallest eigenvector of symmetric
// 3x3 (== V[:,2] of SVD of a PSD matrix), sign-fix, rd. Invalid planes store
// rn = 0, rd = 0 so the downstream update is a no-op.
// refined[b][pl] = {ux,uy,uz,d, rnx,rny,rnz,rd}
// ---------------------------------------------------------------------------
__global__ void __launch_bounds__(64)
k_refine(const float* __restrict__ planes, const float* __restrict__ partial,
         float* __restrict__ refined, int nb1)
{
    const int b  = blockIdx.x;
    const int pl = threadIdx.x;

    float acc[10];
#pragma unroll
    for (int q = 0; q < 10; ++q) acc[q] = 0.f;
    for (int blk = 0; blk < nb1; ++blk) {
        const float* o = partial + (((size_t)b * nb1 + blk) * NPL + pl) * 10;
#pragma unroll
        for (int q = 0; q < 10; ++q) acc[q] += o[q];
    }
    float cnt = acc[0], sx = acc[1], sy = acc[2], sz = acc[3];
    float xx = acc[4], xy = acc[5], xz = acc[6], yy = acc[7], yz = acc[8], zz = acc[9];

    const float* pp = planes + ((size_t)b * NPL + pl) * 4;
    float nx = pp[0], ny = pp[1], nz = pp[2], d = pp[3];
    float nn  = sqrtf(nx * nx + ny * ny + nz * nz);
    float inv = 1.0f / fmaxf(nn, 1e-12f);
    float ux = nx * inv, uy = ny * inv, uz = nz * inv;

    bool valid = (nn >= 1e-6f) && (cnt >= 3.0f);
    float cs = fmaxf(cnt, 1.0f);
    float cx = sx / cs, cy = sy / cs, cz = sz / cs;

    // cov = S2 - S1 S1^T / cnt_safe  (exact for valid planes)
    float a00 = xx - sx * cx, a01 = xy - sx * cy, a02 = xz - sx * cz;
    float a11 = yy - sy * cy, a12 = yz - sy * cz, a22 = zz - sz * cz;
    if (!valid) { a00 = 3.f; a11 = 2.f; a22 = 1.f; a01 = a02 = a12 = 0.f; }

    // trigonometric eigenvalues of symmetric 3x3, smallest eig e3
    float q  = (a00 + a11 + a22) * (1.0f / 3.0f);
    float p1 = a01 * a01 + a02 * a02 + a12 * a12;
    float b00 = a00 - q, b11 = a11 - q, b22 = a22 - q;
    float p2 = b00 * b00 + b11 * b11 + b22 * b22 + 2.0f * p1;
    float p  = sqrtf(fmaxf(p2, 1e-30f) * (1.0f / 6.0f));
    float ip = 1.0f / p;
    float c00 = b00 * ip, c01 = a01 * ip, c02 = a02 * ip;
    float c11 = b11 * ip, c12 = a12 * ip, c22 = b22 * ip;
    float detB = c00 * (c11 * c22 - c12 * c12)
               - c01 * (c01 * c22 - c12 * c02)
               + c02 * (c01 * c12 - c11 * c02);
    float r   = fminf(fmaxf(0.5f * detB, -1.0f), 1.0f);
    float phi = acosf(r) * (1.0f / 3.0f);
    float e3  = q + 2.0f * p * cosf(phi + 2.0943951023931953f); // +2pi/3

    // eigenvector: best cross product of rows of (A - e3 I)
    float m00 = a00 - e3, m11 = a11 - e3, m22 = a22 - e3;
    float v0x = a01 * a12 - a02 * m11, v0y = a02 * a01 - m00 * a12, v0z = m00 * m11 - a01 * a01;
    float v1x = a01 * m22 - a02 * a12, v1y = a02 * a02 - m00 * m22, v1z = m00 * a12 - a01 * a02;
    float v2x = m11 * m22 - a12 * a12, v2y = a12 * a02 - a01 * m22, v2z = a01 * a12 - m11 * a02;
    float n0 = v0x * v0x + v0y * v0y + v0z * v0z;
    float n1 = v1x * v1x + v1y * v1y + v1z * v1z;
    float n2 = v2x * v2x + v2y * v2y + v2z * v2z;
    float vx = v0x, vy = v0y, vz = v0z, bn = n0;
    if (n1 > bn) { vx = v1x; vy = v1y; vz = v1z; bn = n1; }
    if (n2 > bn) { vx = v2x; vy = v2y; vz = v2z; bn = n2; }
    if (bn < 1e-20f) { vx = 0.f; vy = 0.f; vz = 1.f; bn = 1.f; }
    float rs = rsqrtf(bn);
    vx *= rs; vy *= rs; vz *= rs;
    if (vx * ux + vy * uy + vz * uz < 0.f) { vx = -vx; vy = -vy; vz = -vz; }
    float rd = -(cx * vx + cy * vy + cz * vz);

    float* o = refined + ((size_t)b * NPL + pl) * 8;
    o[0] = ux; o[1] = uy; o[2] = uz; o[3] = d;
    if (valid) { o[4] = vx; o[5] = vy; o[6] = vz; o[7] = rd; }
    else       { o[4] = 0.f; o[5] = 0.f; o[6] = 0.f; o[7] = 0.f; }
}

// ---------------------------------------------------------------------------
// Kernel 3: per-point sequential projection. Wave owns 32 points; 8 WMMAs
// give distances to all 64 planes; ballots -> uniform 16-bit "any hit" masks;
// only hit planes (~10/wave) are applied in ascending order with predicated
// per-lane updates (scalar s_load of the 8-float refined record per plane).
// ---------------------------------------------------------------------------
__global__ void __launch_bounds__(256)
k_project(const float* __restrict__ pts, const float* __restrict__ refined,
          float* __restrict__ proj_out, float* __restrict__ disp_out)
{
    const int blocksPerBatch = NPTS / 256;
    const int b     = blockIdx.x / blocksPerBatch;
    const int pbase = (blockIdx.x % blocksPerBatch) * 256;
    const int lane  = threadIdx.x & 31;
    const int wave  = threadIdx.x >> 5;
    const int pi    = pbase + wave * 32 + lane;

    const float* P = pts + ((size_t)b * NPTS + pi) * 3;
    float px = P[0], py = P[1], pz = P[2];

    const int  m  = lane & 15;
    const bool lo = lane < 16;
    float x0 = __shfl(px, m, 32),      y0 = __shfl(py, m, 32),      z0 = __shfl(pz, m, 32);
    float x1 = __shfl(px, m + 16, 32), y1 = __shfl(py, m + 16, 32), z1 = __shfl(pz, m + 16, 32);
    v2f A0, A1;
    A0.x = lo ? x0 : z0;  A0.y = lo ? y0 : 1.0f;
    A1.x = lo ? x1 : z1;  A1.y = lo ? y1 : 1.0f;

    const float* R = refined + (size_t)b * NPL * 8;
    v2f Bf[4];
#pragma unroll
    for (int g = 0; g < 4; ++g) {
        const float* w = R + ((size_t)(g * 16 + m)) * 8;
        float w0 = w[0], w1 = w[1], w2 = w[2], w3 = w[3];
        Bf[g].x = lo ? w0 : w2;
        Bf[g].y = lo ? w1 : w3;
    }

    unsigned hp[4];
#pragma unroll
    for (int g = 0; g < 4; ++g) {
        unsigned acc = 0;
        v8f d0 = wmma_dist(A0, Bf[g]);
#pragma unroll
        for (int j = 0; j < 8; ++j) {
            unsigned bb = (unsigned)__ballot(fabsf(d0[j]) < THRESH);
            acc |= bb | (bb >> 16);
        }
        v8f d1 = wmma_dist(A1, Bf[g]);
#pragma unroll
        for (int j = 0; j < 8; ++j) {
            unsigned bb = (unsigned)__ballot(fabsf(d1[j]) < THRESH);
            acc |= bb | (bb >> 16);
        }
        hp[g] = (unsigned)__builtin_amdgcn_readfirstlane((int)(acc & 0xFFFFu));
    }

    float qx = px, qy = py, qz = pz;
#pragma unroll
    for (int g = 0; g < 4; ++g) {
        unsigned bits = hp[g];
        while (bits) {                               // uniform loop, ascending planes
            int k = (g << 4) + __builtin_ctz(bits);
            bits &= bits - 1;
            const float* w = R + (size_t)k * 8;      // uniform -> s_load_b256
            float u0 = w[0], u1 = w[1], u2 = w[2], u3 = w[3];
            float r0 = w[4], r1 = w[5], r2 = w[6], r3 = w[7];
            bool msk = fabsf(fmaf(px, u0, fmaf(py, u1, fmaf(pz, u2, u3)))) < THRESH;
            float dp = fmaf(qx, r0, fmaf(qy, r1, fmaf(qz, r2, r3)));
            dp = msk ? dp : 0.0f;                    // invalid planes have r=0 anyway
            qx = fmaf(-r0, dp, qx);
            qy = fmaf(-r1, dp, qy);
            qz = fmaf(-r2, dp, qz);
        }
    }

    float* po = proj_out + ((size_t)b * NPTS + pi) * 3;
    float* dq = disp_out + ((size_t)b * NPTS + pi) * 3;
    po[0] = qx;      po[1] = qy;      po[2] = qz;
    dq[0] = qx - px; dq[1] = qy - py; dq[2] = qz - pz;
}

// ---------------------------------------------------------------------------
extern "C" void kernel_launch(void* const* d_in, const int* in_sizes, int n_in,
                              void* d_out, int out_size, void* d_ws, size_t ws_size,
                              hipStream_t stream)
{
    const float* pts    = (const float*)d_in[0];   // [8][262144][3] f32
    const float* planes = (const float*)d_in[1];   // [8][64][4]     f32

    // bp = blocks per batch for kernel 1; each block covers NPTS/bp points and
    // produces 4 partial slots. Shrink until the workspace fits (deterministic).
    int bp = 64;
    while (bp > 1 &&
           ((size_t)NBATCH * (4 * bp) * NPL * 10 + (size_t)NBATCH * NPL * 8) * sizeof(float) > ws_size)
        bp >>= 1;
    const int iters = NPTS / (bp * 64);            // 32-pt steps per wave
    const int nb1   = 4 * bp;                      // partial slots per batch

    float* partial = (float*)d_ws;
    float* refined = partial + (size_t)NBATCH * nb1 * NPL * 10;

    float* proj = (float*)d_out;
    float* disp = proj + (size_t)NBATCH * NPTS * 3;

    hipLaunchKernelGGL(k_plane_stats, dim3(NBATCH * bp), dim3(256), 0, stream,
                       pts, planes, partial, bp, iters);
    hipLaunchKernelGGL(k_refine, dim3(NBATCH), dim3(64), 0, stream,
                       planes, partial, refined, nb1);
    hipLaunchKernelGGL(k_project, dim3(NBATCH * (NPTS / 256)), dim3(256), 0, stream,
                       pts, refined, proj, disp);
}